// RevIN_50216757624895
// MI455X (gfx1250) — compile-verified
//
#include <hip/hip_runtime.h>
#include <math.h>

// RevIN forward (norm mode), fused single kernel for MI455X / gfx1250.
// x: (128, 1024, 128) fp32 = 64MB in / 64MB out -> HBM-bound, ~5.5us floor at 23.3 TB/s.
// One block per batch. Per 256-row chunk: TDM DMA -> LDS (double-buffered, padded rows),
// WMMA f32 16x16x4 row-reduce (sum / sumsq), shfl scan along P, normalize + NaN-fill out.

#define PDIM    1024
#define LDIM    128
#define CHUNK   256
#define NCHUNK  (PDIM / CHUNK)
#define XSTRIDE 132          // 128 + 4 pad floats: conflict-free WMMA A-operand reads
#define NTHREADS 512
#define BUFELEMS (CHUNK * XSTRIDE)

typedef float v2f __attribute__((ext_vector_type(2)));
typedef float v8f __attribute__((ext_vector_type(8)));
typedef unsigned int u32x4 __attribute__((ext_vector_type(4)));
typedef int i32x8 __attribute__((ext_vector_type(8)));
typedef int i32x4 __attribute__((ext_vector_type(4)));

#if defined(__has_builtin)
#if __has_builtin(__builtin_amdgcn_wmma_f32_16x16x4_f32)
#define USE_WMMA_F32X4 1
#endif
#if __has_builtin(__builtin_amdgcn_tensor_load_to_lds)
#define USE_TDM 1
#endif
#endif

#ifdef USE_TDM
// Issue one TDM DMA: 256x128 fp32 tile, row-major global -> LDS with 16B pad per row
// (pad_interval code 6 = every 128 DWORDs, pad_amount code 3 = 4 DWORDs -> 132-float stride).
__device__ static inline void tdm_load_chunk(const float* gsrc, unsigned ldsByteAddr) {
  unsigned long long ga = (unsigned long long)(const void*)gsrc;
  u32x4 g0;
  g0[0] = 1u;                                    // count=1, is_restore=0, gather_mode=0
  g0[1] = ldsByteAddr;                           // lds_addr [63:32]
  g0[2] = (unsigned)(ga & 0xFFFFFFFFu);          // global_addr[31:0]
  g0[3] = (unsigned)((ga >> 32) & 0x01FFFFFFu)   // global_addr[56:32]
        | 0x80000000u;                           // type=2 ("image") in bits [127:126]
  i32x8 g1;
  g1[0] = (2 << 16)                              // data_size = 4 bytes
        | (1 << 20)                              // pad_enable
        | (6 << 22)                              // pad_interval: 128 DWORDs
        | (3 << 25);                             // pad_amount: 4 DWORDs
  g1[1] = (LDIM << 16);                          // atomic_addr=0 | tensor_dim0.lo16 = 128
  g1[2] = (CHUNK << 16);                         // tensor_dim0.hi16=0 | tensor_dim1.lo16 = 256
  g1[3] = (LDIM << 16);                          // tensor_dim1.hi16=0 | tile_dim0 = 128
  g1[4] = CHUNK;                                 // tile_dim1 = 256 | tile_dim2 = 0
  g1[5] = LDIM;                                  // tensor_dim0_stride = 128 elements
  g1[6] = 0;                                     // stride0.hi | tensor_dim1_stride.lo (2D: unused)
  g1[7] = 0;
  i32x4 gz4;
  gz4[0] = 0; gz4[1] = 0; gz4[2] = 0; gz4[3] = 0;  // groups 2/3 unused (<=2D tensor)
  i32x8 gz8;
  gz8[0] = 0; gz8[1] = 0; gz8[2] = 0; gz8[3] = 0;  // extra descriptor words (clang-23
  gz8[4] = 0; gz8[5] = 0; gz8[6] = 0; gz8[7] = 0;  // 6-arg form), zero-filled
  __builtin_amdgcn_tensor_load_to_lds(g0, g1, gz4, gz4, gz8, 0);
}
__device__ static inline void tdm_wait(int allowed) {
#if __has_builtin(__builtin_amdgcn_s_wait_tensorcnt)
  if (allowed) __builtin_amdgcn_s_wait_tensorcnt(1);
  else         __builtin_amdgcn_s_wait_tensorcnt(0);
#else
  if (allowed) asm volatile("s_wait_tensorcnt 1" ::: "memory");
  else         asm volatile("s_wait_tensorcnt 0" ::: "memory");
#endif
}
#endif

__global__ __launch_bounds__(NTHREADS)
void revin_fused_kernel(const float* __restrict__ x, float* __restrict__ out) {
  __shared__ float xs[2 * BUFELEMS];      // double-buffered staged chunks (2 x 256 x 132)
  __shared__ float s_sum[PDIM];           // row sums -> overwritten with mean
  __shared__ float s_sq[PDIM];            // row sum-of-squares -> overwritten with 1/std
  __shared__ float s_cnt[PDIM];           // row NaN counts (cumulative after scan)
  __shared__ float s_wt[24];              // 8 wave totals x 3
  __shared__ float s_carry[3];            // cross-chunk running (S, Q, nan) carry

  const int tid  = threadIdx.x;
  const int lane = tid & 31;
  const int wid  = tid >> 5;

  const size_t bOff = (size_t)blockIdx.x * (size_t)(PDIM * LDIM);
  const float* __restrict__ gx = x + bOff;
  const float4* __restrict__ gx4 = (const float4*)gx;
  float4* __restrict__ gy4 = (float4*)(out + bOff);

  if (tid < 3) s_carry[tid] = 0.0f;

#ifdef USE_TDM
  // Flat LDS address: low 32 bits hold the LDS byte offset (aperture in high bits).
  const unsigned ldsBase = (unsigned)(unsigned long long)(const void*)&xs[0];
  if (wid == 0) tdm_load_chunk(gx, ldsBase);          // pre-issue chunk 0 -> buffer 0
#endif

  for (int c = 0; c < NCHUNK; ++c) {
    const int base = c * CHUNK;
    float* __restrict__ xb = &xs[(c & 1) * BUFELEMS];

#ifdef USE_TDM
    // Pipeline: issue next chunk's DMA into the other buffer, then wait for the
    // current chunk (<=1 outstanding tensor op => the older DMA has completed).
    if ((wid == 0) && (c + 1 < NCHUNK))
      tdm_load_chunk(gx + (size_t)(base + CHUNK) * LDIM,
                     ldsBase + (unsigned)(((c + 1) & 1) * BUFELEMS * 4));
    tdm_wait(c + 1 < NCHUNK);
#else
    // ---------- fallback staging: global -> LDS via VGPRs (coalesced b128) ----------
#pragma unroll
    for (int i = 0; i < (CHUNK * (LDIM / 4)) / NTHREADS; ++i) {
      int flat = i * NTHREADS + tid;          // 0..8191
      int row  = flat >> 5;                   // 32 float4 per row
      int col4 = flat & 31;
      float4 v = gx4[(size_t)(base + row) * (LDIM / 4) + col4];
      *(float4*)&xb[row * XSTRIDE + col4 * 4] = v;
      if ((c + 1 < NCHUNK) && ((flat & 7) == 0)) {  // one prefetch per 128B line
        __builtin_prefetch(&gx4[(size_t)(base + CHUNK + row) * (LDIM / 4) + col4], 0, 1);
      }
    }
#endif
    __syncthreads();

    // ---------- Phase A1: per-row sum / sumsq / nan-count (16-row tile per wave) ----------
    {
      const int tRow = wid * 16;
      const int m    = lane & 15;
      const int koff = (lane >> 4) << 1;      // 0 for lanes 0-15, 2 for lanes 16-31
      const float* rp = &xb[(tRow + m) * XSTRIDE + koff];
      float cnt = 0.0f;
#ifdef USE_WMMA_F32X4
      v8f aS = {0.f, 0.f, 0.f, 0.f, 0.f, 0.f, 0.f, 0.f};
      v8f aQ = {0.f, 0.f, 0.f, 0.f, 0.f, 0.f, 0.f, 0.f};
      const v2f onesB = {1.0f, 1.0f};
#pragma unroll
      for (int k0 = 0; k0 < LDIM; k0 += 4) {
        v2f a = *(const v2f*)(rp + k0);
        float ax = a.x, ay = a.y;
        float n0 = (ax != ax) ? 1.0f : 0.0f;
        float n1 = (ay != ay) ? 1.0f : 0.0f;
        cnt += n0 + n1;
        ax = (ax != ax) ? 0.0f : ax;
        ay = (ay != ay) ? 0.0f : ay;
        v2f av = {ax, ay};
        v2f sq = {ax * ax, ay * ay};
        // D = A(16x4,f32) x ones(4x16) + C : every column of D holds the row sums
        aS = __builtin_amdgcn_wmma_f32_16x16x4_f32(false, av, false, onesB, (short)0, aS, false, false);
        aQ = __builtin_amdgcn_wmma_f32_16x16x4_f32(false, sq, false, onesB, (short)0, aQ, false, false);
      }
      float cA = cnt + __shfl_xor(cnt, 16, 32);
      if (lane < 16) s_cnt[base + tRow + lane] = cA;
      if ((lane & 15) == 0) {
        int rb = base + tRow + ((lane >> 4) << 3);  // lane0 -> rows 0..7, lane16 -> rows 8..15
#pragma unroll
        for (int j = 0; j < 8; ++j) { s_sum[rb + j] = aS[j]; s_sq[rb + j] = aQ[j]; }
      }
#else
      // Fallback: two lanes per row (cols 0-63 / 64-127), shuffle-combine.
      const int half = lane >> 4;
      const float* frp = &xb[(tRow + m) * XSTRIDE + half * 64];
      float s = 0.0f, q = 0.0f;
#pragma unroll
      for (int k = 0; k < 64; k += 4) {
        float4 v = *(const float4*)(frp + k);
        float e[4] = {v.x, v.y, v.z, v.w};
#pragma unroll
        for (int j = 0; j < 4; ++j) {
          float vj  = e[j];
          float bad = (vj != vj) ? 1.0f : 0.0f;
          cnt += bad;
          vj = (vj != vj) ? 0.0f : vj;
          s += vj;
          q += vj * vj;
        }
      }
      s   += __shfl_xor(s, 16, 32);
      q   += __shfl_xor(q, 16, 32);
      cnt += __shfl_xor(cnt, 16, 32);
      if (lane < 16) {
        int rb = base + tRow + lane;
        s_sum[rb] = s; s_sq[rb] = q; s_cnt[rb] = cnt;
      }
#endif
    }
    __syncthreads();

    // ---------- Phase B: inclusive scan of (S,Q,nan) over this chunk + carry ----------
    float tS = 0.0f, tQ = 0.0f, tC = 0.0f;
    if (tid < CHUNK) {
      int gp = base + tid;
      tS = s_sum[gp]; tQ = s_sq[gp]; tC = s_cnt[gp];
#pragma unroll
      for (int d = 1; d < 32; d <<= 1) {
        float oS = __shfl_up(tS, d, 32);
        float oQ = __shfl_up(tQ, d, 32);
        float oC = __shfl_up(tC, d, 32);
        if (lane >= d) { tS += oS; tQ += oQ; tC += oC; }
      }
      if (lane == 31) { s_wt[wid] = tS; s_wt[8 + wid] = tQ; s_wt[16 + wid] = tC; }
    }
    __syncthreads();
    if (tid < CHUNK) {
      float cS = s_carry[0], cQ = s_carry[1], cC = s_carry[2];
      for (int w = 0; w < wid; ++w) { cS += s_wt[w]; cQ += s_wt[8 + w]; cC += s_wt[16 + w]; }
      tS += cS; tQ += cQ; tC += cC;
      int gp = base + tid;
      float count = (float)((gp + 1) * LDIM) - tC;     // cumulative valid count
      float mean  = tS / count;
      float var   = (tQ - 2.0f * mean * tS + count * mean * mean) / count;
      float rstd  = 1.0f / sqrtf(var + 1e-5f);
      s_sum[gp] = mean;                                 // reuse as mean
      s_sq[gp]  = rstd;                                 // reuse as 1/std
      s_cnt[gp] = tC;                                   // cumulative nan count (kept)
    }
    __syncthreads();
    if (tid == CHUNK - 1) { s_carry[0] = tS; s_carry[1] = tQ; s_carry[2] = tC; }

    // ---------- Phase C: normalize + NaN forward-fill, LDS -> global ----------
    {
      const int tRow = wid * 16;
#pragma unroll 4
      for (int r = tRow; r < tRow + 16; ++r) {
        int gp = base + r;
        float mean = s_sum[gp];
        float rstd = s_sq[gp];
        float4 xv = *(const float4*)&xb[r * XSTRIDE + lane * 4];
        float y0 = (xv.x - mean) * rstd;
        float y1 = (xv.y - mean) * rstd;
        float y2 = (xv.z - mean) * rstd;
        float y3 = (xv.w - mean) * rstd;
        bool v0 = (y0 == y0), v1 = (y1 == y1), v2 = (y2 == y2), v3 = (y3 == y3);
        // last valid value within this lane's 4 elements
        float sv = v3 ? y3 : (v2 ? y2 : (v1 ? y1 : (v0 ? y0 : 0.0f)));
        int   sh = (v0 | v1 | v2 | v3) ? 1 : 0;
        // wave inclusive scan with op: b.valid ? b : a
#pragma unroll
        for (int d = 1; d < 32; d <<= 1) {
          float ov = __shfl_up(sv, d, 32);
          int   oh = __shfl_up(sh, d, 32);
          bool take = (lane >= d) && (sh == 0);
          sv = take ? ov : sv;
          sh = take ? oh : sh;
        }
        // exclusive: last valid strictly before this lane (0.0f if none, matching ref)
        float ev = __shfl_up(sv, 1, 32);
        int   eh = __shfl_up(sh, 1, 32);
        float carry = ((lane > 0) && eh) ? ev : 0.0f;
        y0 = v0 ? y0 : carry; carry = y0;
        y1 = v1 ? y1 : carry; carry = y1;
        y2 = v2 ? y2 : carry; carry = y2;
        y3 = v3 ? y3 : carry;
        gy4[(size_t)gp * (LDIM / 4) + lane] = make_float4(y0, y1, y2, y3);
      }
    }
    __syncthreads();  // all reads of this buffer done before its next TDM overwrite
  }
}

extern "C" void kernel_launch(void* const* d_in, const int* in_sizes, int n_in,
                              void* d_out, int out_size, void* d_ws, size_t ws_size,
                              hipStream_t stream) {
  (void)in_sizes; (void)n_in; (void)d_ws; (void)ws_size; (void)out_size;
  const float* x = (const float*)d_in[0];
  float* out = (float*)d_out;
  dim3 grid(128);           // one block per batch
  dim3 block(NTHREADS);     // 16 waves (wave32)
  revin_fused_kernel<<<grid, block, 0, stream>>>(x, out);
}